// Detector_53738630808032
// MI455X (gfx1250) — compile-verified
//
#include <hip/hip_runtime.h>
#include <hip/hip_bf16.h>
#include <stdint.h>

// ---- problem constants (match reference) ----
#define N0 (2048*2048)
#define N1 (1024*1024)
#define N2 (512*512)
#define NTOT (N0+N1+N2)
#define NBINS 16384
#define TOPK 2048          // PRE_NMS_TOPK
#define MAXOUT 2560        // MAX_OUTPUT
#define NGT 1024
#define CANDCAP 4096
#define D2_NMS 64.0f       // NMS_THRESHOLD^2
#define D2_MATCH 144.0f    // MAX_PROPOSAL_OFFSET^2
#define MINSCORE 0.2f

typedef float v2f __attribute__((ext_vector_type(2)));
typedef float v8f __attribute__((ext_vector_type(8)));

__device__ __forceinline__ unsigned ballot32(bool p) {
  return __builtin_amdgcn_ballot_w32(p);
}
// xor-swizzle within the wave: group-of-32 mode, and_mask=0x1f, or=0, xor=M
template<int M>
__device__ __forceinline__ int swz_xor_i(int v) {
  return __builtin_amdgcn_ds_swizzle(v, 0x7C00 | M);
}
template<int M>
__device__ __forceinline__ float swz_xor_f(float v) {
  return __int_as_float(swz_xor_i<M>(__float_as_int(v)));
}

// validity-masked score for flat element e across the 3 pyramid levels
__device__ __forceinline__ float masked_score(int e,
    const float* __restrict__ s0, const float* __restrict__ s1, const float* __restrict__ s2,
    const float* __restrict__ r0, const float* __restrict__ r1, const float* __restrict__ r2) {
  const float* sp; const float* rp; int w, h, o;
  if (e < N0)            { sp = s0; rp = r0; w = 2048; h = 2048; o = e; }
  else if (e < N0 + N1)  { sp = s1; rp = r1; w = 1024; h = 1024; o = e - N0; }
  else                   { sp = s2; rp = r2; w = 512;  h = 512;  o = e - N0 - N1; }
  int i = o / w, j = o - i * w;
  __builtin_prefetch(rp + 2 * o + 4096, 0, 0);   // global_prefetch_b8
  float a0 = rp[2 * o], a1 = rp[2 * o + 1];
  float l0 = (float)i + 0.5f + a0;
  float l1 = (float)j + 0.5f + a1;
  bool valid = (l0 > 0.0f) & (l1 > 0.0f) & (l0 < (float)h) & (l1 < (float)w);
  float s = sp[o];
  return valid ? s : 0.0f;
}

// ---- 1. init scratch ----
__global__ void k_init(unsigned* hist, unsigned* meta, unsigned* first) {
  int t = blockIdx.x * blockDim.x + threadIdx.x;
  if (t < NBINS) hist[t] = 0u;
  if (t < 8)     meta[t] = 0u;
  if (t < NGT)   first[t] = MAXOUT;
}

// ---- 2. score histogram (single streaming pass, ~66MB @ 23.3 TB/s) ----
__global__ void k_hist(const float* __restrict__ s0, const float* __restrict__ s1,
                       const float* __restrict__ s2, const float* __restrict__ r0,
                       const float* __restrict__ r1, const float* __restrict__ r2,
                       unsigned* __restrict__ hist) {
  for (int e = blockIdx.x * blockDim.x + threadIdx.x; e < NTOT; e += gridDim.x * blockDim.x) {
    float s = masked_score(e, s0, s1, s2, r0, r1, r2);
    int b = (int)(s * (float)NBINS);
    b = b < 0 ? 0 : (b > NBINS - 1 ? NBINS - 1 : b);
    atomicAdd(&hist[b], 1u);
  }
}

// ---- 3. pick threshold bin so that >= TOPK candidates survive ----
__global__ void k_thresh(const unsigned* __restrict__ hist, unsigned* meta) {
  if (threadIdx.x == 0 && blockIdx.x == 0) {
    unsigned cum = 0; int b;
    for (b = NBINS - 1; b > 0; --b) { cum += hist[b]; if (cum >= TOPK) break; }
    meta[0] = (unsigned)b;
  }
}

// ---- 4. compact candidates above threshold bin ----
__global__ void k_compact(const float* __restrict__ s0, const float* __restrict__ s1,
                          const float* __restrict__ s2, const float* __restrict__ r0,
                          const float* __restrict__ r1, const float* __restrict__ r2,
                          unsigned* meta, float* __restrict__ candScore,
                          unsigned* __restrict__ candIdx) {
  int tb = (int)meta[0];
  for (int e = blockIdx.x * blockDim.x + threadIdx.x; e < NTOT; e += gridDim.x * blockDim.x) {
    float s = masked_score(e, s0, s1, s2, r0, r1, r2);
    int b = (int)(s * (float)NBINS);
    b = b < 0 ? 0 : (b > NBINS - 1 ? NBINS - 1 : b);
    if (b >= tb) {
      unsigned pos = atomicAdd(&meta[1], 1u);
      if (pos < CANDCAP) { candScore[pos] = s; candIdx[pos] = (unsigned)e; }
    }
  }
}

// ---- 5. single-workgroup bitonic sort (descending score, then ascending index) ----
__global__ void __launch_bounds__(1024)
k_sort(const unsigned* meta, const float* __restrict__ candScore,
       const unsigned* __restrict__ candIdx, float* __restrict__ sortedScore,
       unsigned* __restrict__ sortedIdx) {
  __shared__ unsigned long long keys[CANDCAP];   // 32 KB of the 320 KB LDS
  unsigned n = meta[1]; if (n > CANDCAP) n = CANDCAP;
  for (int i = threadIdx.x; i < CANDCAP; i += 1024) {
    unsigned long long k = 0ull;
    if (i < (int)n) {
      unsigned sb = __float_as_uint(candScore[i]);
      k = ((unsigned long long)sb << 32) | (unsigned long long)(0xFFFFFFFFu - candIdx[i]);
    }
    keys[i] = k;
  }
  __syncthreads();
  for (int k = 2; k <= CANDCAP; k <<= 1) {
    for (int j = k >> 1; j > 0; j >>= 1) {
      for (int i = threadIdx.x; i < CANDCAP; i += 1024) {
        int ixj = i ^ j;
        if (ixj > i) {
          bool dirDesc = ((i & k) == 0);
          unsigned long long a = keys[i], b = keys[ixj];
          bool doSwap = dirDesc ? (a < b) : (a > b);
          if (doSwap) { keys[i] = b; keys[ixj] = a; }
        }
      }
      __syncthreads();
    }
  }
  for (int i = threadIdx.x; i < TOPK; i += 1024) {
    unsigned long long k = keys[i];
    sortedScore[i] = __uint_as_float((unsigned)(k >> 32));
    sortedIdx[i]   = 0xFFFFFFFFu - (unsigned)(k & 0xFFFFFFFFull);
  }
}

// ---- 6. gather scaled locations for the sorted top-K ----
__global__ void k_gather(const unsigned* __restrict__ sortedIdx,
                         const float* __restrict__ r0, const float* __restrict__ r1,
                         const float* __restrict__ r2, float* __restrict__ sortedLoc) {
  int t = blockIdx.x * blockDim.x + threadIdx.x;
  if (t >= TOPK) return;
  unsigned idx = sortedIdx[t];
  float lx = 1e30f, ly = 1e30f;    // padded entries: far away, score 0 -> never kept
  if (idx < (unsigned)NTOT) {
    const float* rp; int w, o; float sc;
    if (idx < (unsigned)N0)            { rp = r0; w = 2048; o = (int)idx;            sc = 1.f; }
    else if (idx < (unsigned)(N0+N1))  { rp = r1; w = 1024; o = (int)idx - N0;       sc = 2.f; }
    else                               { rp = r2; w = 512;  o = (int)idx - N0 - N1;  sc = 4.f; }
    int i = o / w, j = o - i * w;
    lx = ((float)i + 0.5f + rp[2 * o])     * sc;
    ly = ((float)j + 0.5f + rp[2 * o + 1]) * sc;
  }
  sortedLoc[2 * t] = lx; sortedLoc[2 * t + 1] = ly;
}

// ---- 7. WMMA suppression bit-matrix: susp[i][j] = (d2(i,j) < 64) ----
// d2 = |pi|^2 + |pj|^2 - 2 pi.pj  via one F32 16x16x4 WMMA per 16x16 tile.
// Each wave: 16 rows x 32 cols (two WMMAs), ballot-packed into one u32/row.
__global__ void k_susp(const float* __restrict__ sortedLoc, unsigned* __restrict__ susp) {
  int wave = (int)((blockIdx.x * blockDim.x + threadIdx.x) >> 5);  // 0..8191
  int lane = (int)(threadIdx.x & 31);
  int ti = wave >> 6;      // 0..127 row tile
  int tj = wave & 63;      // 0..63 32-column group
  int i0 = ti * 16, j0 = tj * 32;
  int lm = lane & 15;
  bool hiK = lane >= 16;   // lanes 16..31 supply K=2,3 (|p|^2, 1)

  float px = sortedLoc[2 * (i0 + lm)], py = sortedLoc[2 * (i0 + lm) + 1];
  v2f a; a.x = hiK ? (px * px + py * py) : px;
         a.y = hiK ? 1.0f : py;

  float qx0 = sortedLoc[2 * (j0 + lm)],      qy0 = sortedLoc[2 * (j0 + lm) + 1];
  float qx1 = sortedLoc[2 * (j0 + 16 + lm)], qy1 = sortedLoc[2 * (j0 + 16 + lm) + 1];
  v2f b0; b0.x = hiK ? 1.0f : (-2.0f * qx0);
          b0.y = hiK ? (qx0 * qx0 + qy0 * qy0) : (-2.0f * qy0);
  v2f b1; b1.x = hiK ? 1.0f : (-2.0f * qx1);
          b1.y = hiK ? (qx1 * qx1 + qy1 * qy1) : (-2.0f * qy1);

  v8f c0 = {}, c1 = {};
  c0 = __builtin_amdgcn_wmma_f32_16x16x4_f32(false, a, false, b0, (short)0, c0, false, false);
  c1 = __builtin_amdgcn_wmma_f32_16x16x4_f32(false, a, false, b1, (short)0, c1, false, false);

  unsigned mA[8], mB[8];
#pragma unroll
  for (int k = 0; k < 8; ++k) mA[k] = ballot32(c0[k] < D2_NMS);
#pragma unroll
  for (int k = 0; k < 8; ++k) mB[k] = ballot32(c1[k] < D2_NMS);

  if (lane < 16) {               // lane r writes the word for row i0+r
    int r = lane, k = r & 7;
    unsigned w;
    if (r < 8) w = (mA[k] & 0xFFFFu) | ((mB[k] & 0xFFFFu) << 16);
    else       w = (mA[k] >> 16)     | (mB[k] & 0xFFFF0000u);
    susp[(unsigned)(i0 + r) * 64u + (unsigned)tj] = w;
  }
}

// ---- 8. greedy NMS scan: single wave walks the bit-matrix sequentially.
// Rows are streamed ahead with gfx1250 async global->LDS copies (ASYNCcnt),
// double-buffered in LDS: row i+1 is in flight while row i is tested.
__global__ void k_nms(const float* __restrict__ sortedScore, const unsigned* __restrict__ susp,
                      unsigned* __restrict__ sel, unsigned* meta) {
  __shared__ __align__(16) unsigned long long rowbuf[2][32];  // 2 x 256B rows
  int lane = (int)(threadIdx.x & 31);
  unsigned kw0 = 0u, kw1 = 0u;   // this lane owns kept-bitmask words 2*lane, 2*lane+1
  unsigned cnt = 0u;

  // prologue: async-load row 0 into buffer 0 (each lane pulls its 8B slice)
  {
    unsigned l = (unsigned)(uintptr_t)(&rowbuf[0][lane]);      // low 32 bits = LDS offset
    unsigned long long g = (unsigned long long)(uintptr_t)susp + (unsigned long long)lane * 8ull;
    asm volatile("global_load_async_to_lds_b64 %0, %1, off" :: "v"(l), "v"(g) : "memory");
  }

  for (int i = 0; i < TOPK; ++i) {
    if (i + 1 < TOPK) {
      unsigned l = (unsigned)(uintptr_t)(&rowbuf[(i + 1) & 1][lane]);
      unsigned long long g = (unsigned long long)(uintptr_t)susp
                           + (unsigned long long)(i + 1) * 256ull
                           + (unsigned long long)lane * 8ull;
      asm volatile("global_load_async_to_lds_b64 %0, %1, off" :: "v"(l), "v"(g) : "memory");
      // async loads complete in order: <=1 outstanding means row i has landed
      asm volatile("s_wait_asynccnt 0x1" ::: "memory");
    } else {
      asm volatile("s_wait_asynccnt 0x0" ::: "memory");
    }
    unsigned long long rr = rowbuf[i & 1][lane];
    unsigned r0 = (unsigned)rr;           // susp word 2*lane
    unsigned r1 = (unsigned)(rr >> 32);   // susp word 2*lane+1
    bool hit = ((r0 & kw0) | (r1 & kw1)) != 0u;
    bool sup = ballot32(hit) != 0u;                  // uniform
    float s = sortedScore[i];
    bool keep = (s >= MINSCORE) && !sup;             // uniform
    if (keep) {
      unsigned w = (unsigned)i >> 5, b = (unsigned)i & 31u;
      if (w == 2u * (unsigned)lane)      kw0 |= (1u << b);
      if (w == 2u * (unsigned)lane + 1u) kw1 |= (1u << b);
      if (lane == 0) sel[cnt] = (unsigned)i;
      ++cnt;
    }
  }
  if (lane == 0) meta[2] = cnt;
}

// ---- 9. emit pred_scores / pred_locations (with -1 fill) ----
__global__ void k_emit(const unsigned* meta, const unsigned* __restrict__ sel,
                       const float* __restrict__ sortedScore, const float* __restrict__ sortedLoc,
                       float* __restrict__ outScores, float* __restrict__ outLocs) {
  int t = blockIdx.x * blockDim.x + threadIdx.x;
  if (t >= MAXOUT) return;
  unsigned cnt = meta[2];
  if (t < (int)cnt) {
    unsigned i = sel[t];
    outScores[t] = sortedScore[i];
    outLocs[2 * t]     = sortedLoc[2 * i];
    outLocs[2 * t + 1] = sortedLoc[2 * i + 1];
  } else {
    outScores[t] = -1.0f;
    outLocs[2 * t] = -1.0f;
    outLocs[2 * t + 1] = -1.0f;
  }
}

// ---- 10. WMMA GT matching: per pred row, nearest GT (argmin) within 12 ----
__global__ void k_match(const float* __restrict__ predLoc, const float* __restrict__ gt,
                        int* __restrict__ matched) {
  int wave = (int)((blockIdx.x * blockDim.x + threadIdx.x) >> 5);  // 0..159
  int lane = (int)(threadIdx.x & 31);
  if (wave >= MAXOUT / 16) return;      // uniform per wave
  int m0 = wave * 16;
  int lm = lane & 15;
  bool hiK = lane >= 16;

  float px = predLoc[2 * (m0 + lm)], py = predLoc[2 * (m0 + lm) + 1];
  v2f a; a.x = hiK ? (px * px + py * py) : px;
         a.y = hiK ? 1.0f : py;

  float best[8]; int bidx[8];
#pragma unroll
  for (int k = 0; k < 8; ++k) { best[k] = 3.4e38f; bidx[k] = 0; }

  for (int g0 = 0; g0 < NGT; g0 += 16) {
    float qx = gt[2 * (g0 + lm)], qy = gt[2 * (g0 + lm) + 1];
    v2f b; b.x = hiK ? 1.0f : (-2.0f * qx);
           b.y = hiK ? (qx * qx + qy * qy) : (-2.0f * qy);
    v8f c = {};
    c = __builtin_amdgcn_wmma_f32_16x16x4_f32(false, a, false, b, (short)0, c, false, false);
    int gi = g0 + lm;
#pragma unroll
    for (int k = 0; k < 8; ++k) {
      float d = c[k];
      if (d < best[k]) { best[k] = d; bidx[k] = gi; }   // strict: earlier tile wins ties
    }
  }
  // reduce (min, first-index) across the 16 lanes that share the same rows
#pragma unroll
  for (int k = 0; k < 8; ++k) {
    float d = best[k]; int gi = bidx[k];
    { float od = swz_xor_f<8>(d); int oi = swz_xor_i<8>(gi);
      if (od < d || (od == d && oi < gi)) { d = od; gi = oi; } }
    { float od = swz_xor_f<4>(d); int oi = swz_xor_i<4>(gi);
      if (od < d || (od == d && oi < gi)) { d = od; gi = oi; } }
    { float od = swz_xor_f<2>(d); int oi = swz_xor_i<2>(gi);
      if (od < d || (od == d && oi < gi)) { d = od; gi = oi; } }
    { float od = swz_xor_f<1>(d); int oi = swz_xor_i<1>(gi);
      if (od < d || (od == d && oi < gi)) { d = od; gi = oi; } }
    best[k] = d; bidx[k] = gi;
  }
  if (lm == 0) {   // lanes 0 (rows m0..m0+7) and 16 (rows m0+8..m0+15)
    int rowBase = m0 + (hiK ? 8 : 0);
#pragma unroll
    for (int k = 0; k < 8; ++k)
      matched[rowBase + k] = (best[k] < D2_MATCH) ? bidx[k] : -1;
  }
}

// ---- 11. first (highest-score) matching prediction per GT ----
__global__ void k_first(const int* __restrict__ matched, unsigned* __restrict__ first) {
  int t = blockIdx.x * blockDim.x + threadIdx.x;
  if (t >= MAXOUT) return;
  int g = matched[t];
  if (g >= 0) atomicMin(&first[g], (unsigned)t);
}

// ---- 12. training locations ----
__global__ void k_train(const unsigned* __restrict__ first, const float* __restrict__ predLoc,
                        const float* __restrict__ gt, float* __restrict__ outTrain) {
  int g = blockIdx.x * blockDim.x + threadIdx.x;
  if (g >= NGT) return;
  unsigned f = first[g];
  if (f >= MAXOUT) { outTrain[2 * g] = gt[2 * g];       outTrain[2 * g + 1] = gt[2 * g + 1]; }
  else             { outTrain[2 * g] = predLoc[2 * f];  outTrain[2 * g + 1] = predLoc[2 * f + 1]; }
}

extern "C" void kernel_launch(void* const* d_in, const int* in_sizes, int n_in,
                              void* d_out, int out_size, void* d_ws, size_t ws_size,
                              hipStream_t stream) {
  const float* s0 = (const float*)d_in[0];
  const float* s1 = (const float*)d_in[1];
  const float* s2 = (const float*)d_in[2];
  const float* r0 = (const float*)d_in[3];
  const float* r1 = (const float*)d_in[4];
  const float* r2 = (const float*)d_in[5];
  const float* gt = (const float*)d_in[6];

  float* out = (float*)d_out;
  float* outScores = out;                       // [2560]
  float* outLocs   = out + MAXOUT;              // [2560,2]
  float* outTrain  = out + MAXOUT + 2 * MAXOUT; // [1024,2]

  char* ws = (char*)d_ws;
  size_t off = 0;
  auto alloc = [&](size_t bytes) -> void* {
    void* p = ws + off; off = (off + bytes + 255) & ~(size_t)255; return p;
  };
  unsigned* hist        = (unsigned*)alloc(NBINS * 4);
  unsigned* meta        = (unsigned*)alloc(64);
  float*    candScore   = (float*)alloc(CANDCAP * 4);
  unsigned* candIdx     = (unsigned*)alloc(CANDCAP * 4);
  float*    sortedScore = (float*)alloc(TOPK * 4);
  unsigned* sortedIdx   = (unsigned*)alloc(TOPK * 4);
  float*    sortedLoc   = (float*)alloc(TOPK * 8);
  unsigned* susp        = (unsigned*)alloc((size_t)TOPK * 64 * 4);  // 512 KB
  unsigned* sel         = (unsigned*)alloc(MAXOUT * 4);
  int*      matched     = (int*)alloc(MAXOUT * 4);
  unsigned* first       = (unsigned*)alloc(NGT * 4);

  k_init<<<64, 256, 0, stream>>>(hist, meta, first);
  k_hist<<<2048, 256, 0, stream>>>(s0, s1, s2, r0, r1, r2, hist);
  k_thresh<<<1, 1, 0, stream>>>(hist, meta);
  k_compact<<<2048, 256, 0, stream>>>(s0, s1, s2, r0, r1, r2, meta, candScore, candIdx);
  k_sort<<<1, 1024, 0, stream>>>(meta, candScore, candIdx, sortedScore, sortedIdx);
  k_gather<<<(TOPK + 255) / 256, 256, 0, stream>>>(sortedIdx, r0, r1, r2, sortedLoc);
  k_susp<<<1024, 256, 0, stream>>>(sortedLoc, susp);      // 8192 waves, 2 WMMAs each
  k_nms<<<1, 32, 0, stream>>>(sortedScore, susp, sel, meta);
  k_emit<<<(MAXOUT + 255) / 256, 256, 0, stream>>>(meta, sel, sortedScore, sortedLoc,
                                                   outScores, outLocs);
  k_match<<<(MAXOUT / 16) * 32 / 256, 256, 0, stream>>>(outLocs, gt, matched); // 160 waves
  k_first<<<(MAXOUT + 255) / 256, 256, 0, stream>>>(matched, first);
  k_train<<<(NGT + 255) / 256, 256, 0, stream>>>(first, outLocs, gt, outTrain);
}